// Warper_85134841741391
// MI455X (gfx1250) — compile-verified
//
#include <hip/hip_runtime.h>

// Forward-warp (bilinear splat) for MI455X / gfx1250.
// Pass 1: zero accumulators (wm in d_ws, bf in d_out).
// Pass 2: per-source-pixel scatter with global_atomic_add_f32 (L2-resident on
//         MI455X: 50 MB of accumulators << 192 MB L2).
// Pass 3: streaming normalize in place.

#define BS   8
#define H    544
#define W    960
#define HW   (H * W)
#define NTOT (BS * HW)
#define EPS  0.001f

__global__ void warper_zero_f4(float4* __restrict__ p, int n4) {
    int i = blockIdx.x * blockDim.x + threadIdx.x;
    if (i < n4) p[i] = make_float4(0.f, 0.f, 0.f, 0.f);
}

__global__ void warper_splat(const float2* __restrict__ flows,
                             float* __restrict__ wm,     // [BS*HW]
                             float* __restrict__ bf) {   // [BS*HW*2] (== d_out)
    int i = blockIdx.x * blockDim.x + threadIdx.x;
    if (i >= NTOT) return;

    // gfx1250: global_prefetch_b8 a few cachelines ahead of the stream.
    __builtin_prefetch((const void*)(flows + i + 2048), 0, 0);

    float2 f = flows[i];
    int b  = i / HW;
    int n  = i - b * HW;
    int py = n / W;
    int px = n - py * W;

    float lf = floorf(f.x);
    float tf = floorf(f.y);
    bool m = (lf >= 0.f) && (tf >= 0.f) &&
             (lf + 1.f < (float)W) && (tf + 1.f < (float)H);

    // Masked pixels: l=r=t=b=x=y=0 -> all four deltas collapse to EPS and all
    // four corner indices collapse to 0 (reference semantics, must keep!).
    float fx = m ? (f.x - lf) : 0.f;           // x - l   (exact in f32)
    float fy = m ? (f.y - tf) : 0.f;           // y - t
    float dx0 = fx + EPS;                      // |x - l + eps|
    float dy0 = fy + EPS;                      // |y - t + eps|
    float dx1 = m ? ((1.f - fx) + EPS) : EPS;  // |r - x + eps|
    float dy1 = m ? ((1.f - fy) + EPS) : EPS;  // |b - y + eps|

    int li = m ? (int)lf : 0;
    int ti = m ? (int)tf : 0;
    int ri = m ? (li + 1) : 0;
    int bi = m ? (ti + 1) : 0;

    // Shifts use the ORIGINAL (unmasked) flow.
    float sx = (float)px - f.x;
    float sy = (float)py - f.y;

    float w_lt = dx1 * dy1;   // -> (t, l)
    float w_rt = dx0 * dy1;   // -> (t, r)
    float w_lb = dx1 * dy0;   // -> (b, l)
    float w_rb = dx0 * dy0;   // -> (b, r)

    int i_tl = ti * W + li;
    int i_tr = ti * W + ri;
    int i_bl = bi * W + li;
    int i_br = bi * W + ri;

    float* wmb = wm + b * HW;
    float* bfb = bf + (size_t)b * HW * 2;

    atomicAdd(wmb + i_tl,         w_lt);
    atomicAdd(bfb + 2 * i_tl,     sx * w_lt);
    atomicAdd(bfb + 2 * i_tl + 1, sy * w_lt);

    atomicAdd(wmb + i_tr,         w_rt);
    atomicAdd(bfb + 2 * i_tr,     sx * w_rt);
    atomicAdd(bfb + 2 * i_tr + 1, sy * w_rt);

    atomicAdd(wmb + i_bl,         w_lb);
    atomicAdd(bfb + 2 * i_bl,     sx * w_lb);
    atomicAdd(bfb + 2 * i_bl + 1, sy * w_lb);

    atomicAdd(wmb + i_br,         w_rb);
    atomicAdd(bfb + 2 * i_br,     sx * w_rb);
    atomicAdd(bfb + 2 * i_br + 1, sy * w_rb);
}

__global__ void warper_normalize(const float* __restrict__ wm,
                                 float2* __restrict__ out) {  // in place on d_out
    int i = blockIdx.x * blockDim.x + threadIdx.x;
    if (i >= NTOT) return;

    int b  = i / HW;
    int n  = i - b * HW;
    int py = n / W;
    int px = n - py * W;

    float wv = wm[i];
    float2 v = out[i];
    float2 o;
    if (wv == 0.f) {
        o.x = -2.f; o.y = -2.f;
    } else {
        float qx = v.x / wv;
        float qy = v.y / wv;
        if (!isfinite(qx)) qx = 0.f;
        if (!isfinite(qy)) qy = 0.f;
        o.x = qx + (float)px;
        o.y = qy + (float)py;
    }
    out[i] = o;
}

extern "C" void kernel_launch(void* const* d_in, const int* in_sizes, int n_in,
                              void* d_out, int out_size, void* d_ws, size_t ws_size,
                              hipStream_t stream) {
    (void)in_sizes; (void)n_in; (void)out_size; (void)ws_size;

    const float2* flows = (const float2*)d_in[0];  // [BS,H,W,2] float32
    float*  wm = (float*)d_ws;                     // [BS*HW] weight map (16.7 MB)
    float*  bf = (float*)d_out;                    // [BS*HW*2] accum -> final out

    const int T = 256;  // 8 wave32 per block

    // Zero wm (NTOT floats) and bf (2*NTOT floats) with b128 stores.
    warper_zero_f4<<<(NTOT / 4 + T - 1) / T, T, 0, stream>>>((float4*)wm, NTOT / 4);
    warper_zero_f4<<<(NTOT / 2 + T - 1) / T, T, 0, stream>>>((float4*)bf, NTOT / 2);

    // Scatter splat: 12 L2-resident f32 atomics per source pixel.
    warper_splat<<<(NTOT + T - 1) / T, T, 0, stream>>>(flows, wm, bf);

    // Streaming normalize in place.
    warper_normalize<<<(NTOT + T - 1) / T, T, 0, stream>>>(wm, (float2*)bf);
}